// SPLayer_88064009437611
// MI455X (gfx1250) — compile-verified
//
#include <hip/hip_runtime.h>
#include <stdint.h>

#define NWAVES 8
#define BIGF 1e9f

__device__ __forceinline__ float shup(float x, int d) { return __shfl_up(x, (unsigned)d, 32); }
__device__ __forceinline__ float shdn(float x, int d) { return __shfl_down(x, (unsigned)d, 32); }
__device__ __forceinline__ float sq(float x) { return x * x; }

// softmin2(a,b) = -log(e^-a + e^-b); exact pass-through when one arg is BIG
__device__ __forceinline__ float softmin2(float a, float b) {
    float d = fabsf(a - b);
    return fminf(a, b) - log1pf(__expf(-d));
}
__device__ __forceinline__ float softmin3(float a, float b, float c) {
    float m = fminf(fminf(a, b), c);
    float s = __expf(m - a) + __expf(m - b) + __expf(m - c);
    return m - __logf(s);
}

// One wave (32 lanes) per image; lane j handles column j of the 32x32 grid.
__global__ __launch_bounds__(NWAVES * 32) void sp_dp_kernel(
    const float* __restrict__ image, float* __restrict__ out_v,
    float* __restrict__ out_E, float* __restrict__ out_vh, int nimg)
{
    __shared__ __align__(16) float sp[NWAVES * 1024];  // image, then sigmoid(image)
    __shared__ __align__(16) float sv[NWAVES * 1024];  // soft DP values v[i][j]

    const int lane = threadIdx.x & 31;
    const int w    = threadIdx.x >> 5;
    const int img  = blockIdx.x * NWAVES + w;
    if (img >= nimg) return;
    const int base = w * 1024;

    // ---- Stage this wave's image into LDS via CDNA5 async-to-LDS (ASYNCcnt) ----
    {
        uint32_t lbase = (uint32_t)(uintptr_t)(&sp[base]);   // low 32 bits = LDS byte addr
        const char* g  = (const char*)(image + (size_t)img * 1024);
#pragma unroll
        for (int q = 0; q < 8; ++q) {
            uint32_t la = lbase + (uint32_t)(q * 512 + lane * 16);
            uint64_t ga = (uint64_t)(uintptr_t)(g + q * 512 + lane * 16);
            asm volatile("global_load_async_to_lds_b128 %0, %1, off"
                         :: "v"(la), "v"(ga) : "memory");
        }
        asm volatile("s_wait_asynccnt 0x0" ::: "memory");
    }

    // ================= forward (soft + hard fused) =================
    // row 0: v0[j] = sum_{k<j} tE[0][k]  (exclusive prefix sum, Kogge-Stone)
    float x0 = sp[base + lane];
    float p_r = 1.0f / (1.0f + __expf(-x0));   // p row 0
    sp[base + lane] = p_r;                      // overwrite raw with sigmoid
    float v_prev, hv_prev;
    {
        float pdn = shdn(p_r, 1);
        float te0 = (lane < 31) ? sq(p_r + pdn) : 0.0f;
        float s = te0;
#pragma unroll
        for (int d = 1; d < 32; d <<= 1) { float t = shup(s, d); if (lane >= d) s += t; }
        float v0 = shup(s, 1); if (lane == 0) v0 = 0.0f;
        sv[base + lane] = v0;
        v_prev = v0; hv_prev = v0;
    }

    for (int i = 1; i < 32; ++i) {
        float xr = sp[base + i * 32 + lane];
        float p  = 1.0f / (1.0f + __expf(-xr));   // p row i
        sp[base + i * 32 + lane] = p;
        float p_dn = shdn(p, 1);                  // p[i][j+1]
        float p_up = shup(p, 1);                  // p[i][j-1]
        // theta rows (p_r = p[i-1])
        float tse = (lane < 31) ? sq(p_r + p_dn) : BIGF;  // tSE[i-1][j]
        float ts  = sq(p_r + p);                          // tS [i-1][j]
        float tsw = (lane > 0)  ? sq(p_r + p_up) : BIGF;  // tSW[i-1][j]
        float te  = (lane < 31) ? sq(p   + p_dn) : BIGF;  // tE [i][j]
        float te_in = shup(te, 1); if (lane == 0) te_in = BIGF;  // tE[i][j-1]
        // cross-row candidates
        float a = shup(v_prev + tse, 1); if (lane == 0)  a = BIGF;
        float b = v_prev + ts;
        float c = shdn(v_prev + tsw, 1); if (lane == 31) c = BIGF;
        float u = softmin3(a, b, c);
        float ah = shup(hv_prev + tse, 1); if (lane == 0)  ah = BIGF;
        float bh = hv_prev + ts;
        float ch = shdn(hv_prev + tsw, 1); if (lane == 31) ch = BIGF;
        float uh = fminf(fminf(ah, bh), ch);
        // in-row recurrence v_j = min2(u_j, v_{j-1}+te_in_j): affine-map scan
        float A = u,  M = te_in;       // soft: w -> e^-A + e^-M * w (log domain)
        float Ah = uh, Mh = te_in;     // hard: w -> min(Ah, Mh + w)
#pragma unroll
        for (int d = 1; d < 32; d <<= 1) {
            float Alo = shup(A, d),  Mlo  = shup(M, d);
            float Blo = shup(Ah, d), MHlo = shup(Mh, d);
            if (lane >= d) {
                A  = softmin2(A, M + Alo);  M  += Mlo;
                Ah = fminf(Ah, Mh + Blo);   Mh += MHlo;
            }
        }
        // initial carry contributes via te_in[0]=BIG (weight 0) -> v = A exactly
        sv[base + i * 32 + lane] = A;
        v_prev = A; hv_prev = Ah; p_r = p;
    }
    if (lane == 31) { out_v[img] = v_prev; out_vh[img] = hv_prev; }

    // ================= backward (analytic grad of soft DP) =================
    float cSE = 0.0f, cS = 0.0f, cSW = 0.0f;            // SE/S/SW grads for row r-1
    float bg  = (lane == 31) ? 1.0f : 0.0f;             // base adjoint for row 31
    float v_r = v_prev;                                 // v row 31 ; p_r = p row 31
    float* Ebase = out_E + (size_t)img * 4096;

    for (int r = 31; r >= 1; --r) {
        float p_p = sp[base + (r - 1) * 32 + lane];
        float v_p = sv[base + (r - 1) * 32 + lane];
        float prdn = shdn(p_r, 1), prup = shup(p_r, 1);
        float tse = (lane < 31) ? sq(p_p + prdn) : BIGF;
        float ts  = sq(p_p + p_r);
        float tsw = (lane > 0)  ? sq(p_p + prup) : BIGF;
        float te  = (lane < 31) ? sq(p_r + prdn) : BIGF;
        float te_in = shup(te, 1); if (lane == 0) te_in = BIGF;
        float a = shup(v_p + tse, 1); if (lane == 0)  a = BIGF;
        float b = v_p + ts;
        float c = shdn(v_p + tsw, 1); if (lane == 31) c = BIGF;
        float u = softmin3(a, b, c);
        // softmin2 weights at node (r,j): pU + pE = 1
        float vjm = shup(v_r, 1);
        float pE = __expf(v_r - (vjm + te_in));  // lane0: exp(-BIG) = 0
        float pU = __expf(v_r - u);
        // g_j = bg_j + pE_{j+1} * g_{j+1} : right-to-left affine scan
        float A = bg;
        float M = shdn(pE, 1); if (lane == 31) M = 0.0f;
#pragma unroll
        for (int d = 1; d < 32; d <<= 1) {
            float Ahi = shdn(A, d), Mhi = shdn(M, d);
            if (lane + d < 32) { A += M * Ahi; M *= Mhi; }
        }
        float g = A;
        // East grad for row r: dtE[r][j] = g[r][j+1]*pE[r][j+1]
        float gE = shdn(g * pE, 1); if (lane == 31) gE = 0.0f;
        *(float4*)(Ebase + r * 128 + lane * 4) = make_float4(gE, cSE, cS, cSW);
        // softmin3 weights -> grads into row r-1 edges & base adjoint
        float gu = g * pU;
        float wa = gu * __expf(u - a);
        float wb = gu * __expf(u - b);
        float wc = gu * __expf(u - c);
        cSE = shdn(wa, 1); if (lane == 31) cSE = 0.0f;   // dtSE[r-1][j]
        cS  = wb;                                        // dtS [r-1][j]
        cSW = shup(wc, 1); if (lane == 0)  cSW = 0.0f;   // dtSW[r-1][j]
        bg  = cSE + cS + cSW;                            // d v[r-1][j]
        v_r = v_p; p_r = p_p;
    }
    // row 0: g[0][j] = suffix-sum of bg; dtE[0][j] = g[0][j+1]
    float G = bg;
#pragma unroll
    for (int d = 1; d < 32; d <<= 1) { float Gh = shdn(G, d); if (lane + d < 32) G += Gh; }
    float gE0 = shdn(G, 1); if (lane == 31) gE0 = 0.0f;
    *(float4*)(Ebase + lane * 4) = make_float4(gE0, cSE, cS, cSW);
}

extern "C" void kernel_launch(void* const* d_in, const int* in_sizes, int n_in,
                              void* d_out, int out_size, void* d_ws, size_t ws_size,
                              hipStream_t stream) {
    (void)n_in; (void)out_size; (void)d_ws; (void)ws_size;
    const float* image = (const float*)d_in[0];
    int nimg = in_sizes[0] / 1024;          // B images of 32x32
    float* out    = (float*)d_out;
    float* out_v  = out;                    // [B]
    float* out_E  = out + nimg;             // [B,1024,4]
    float* out_vh = out + nimg + (size_t)nimg * 4096;  // [B]
    int blocks = (nimg + NWAVES - 1) / NWAVES;
    sp_dp_kernel<<<blocks, NWAVES * 32, 0, stream>>>(image, out_v, out_E, out_vh, nimg);
}